// loss_81535659148068
// MI455X (gfx1250) — compile-verified
//
#include <hip/hip_runtime.h>
#include <math.h>

// ---------------------------------------------------------------------------
// Loss kernel for MI455X (gfx1250, wave32).
//  out = sum_e max(1 - ||out[i_e]-out[j_e]+eps||, 0) - (log(0.1)+sum_c log(bincount(argmax)_c))
//
// Edge term uses V_WMMA_F32_16X16X4_F32: per 16-edge group, the 16x16 Gram
// matrix D = A*A^T is accumulated over K (64 channels / 4 per WMMA); its
// diagonal holds the squared distances. For f32 WMMA the A(16x4) and B(4x16)
// register layouts coincide (K -> {vgpr, lane-half}, row/col -> lane-in-half),
// so passing the same regs as A and B computes A*A^T with zero shuffles.
// ---------------------------------------------------------------------------

typedef __attribute__((ext_vector_type(2))) float v2f;
typedef __attribute__((ext_vector_type(8))) float v8f;

#define MARGIN 1.0f
#define EPSF   1e-6f
#define CCH    64          // channels
#define EDGE_BLOCKS 1024   // blocks for edge kernel (partials count)

// ---- zero the 64 class counters (d_ws is not re-poisoned between calls) ----
__global__ void init_counts_kernel(int* __restrict__ counts) {
    if (threadIdx.x < CCH) counts[threadIdx.x] = 0;
}

// ---- per-node argmax (first-max semantics) + bincount via LDS + atomics ----
__global__ __launch_bounds__(256) void argmax_count_kernel(
    const float* __restrict__ out, int N, int* __restrict__ counts) {
    __shared__ int lcounts[CCH];
    if (threadIdx.x < CCH) lcounts[threadIdx.x] = 0;
    __syncthreads();

    int t = blockIdx.x * blockDim.x + threadIdx.x;
    if (t < N) {
        const float* row = out + (long long)t * CCH;
        float best = -3.402823466e+38f;
        int bi = 0;
        #pragma unroll
        for (int c4 = 0; c4 < CCH / 4; ++c4) {
            float4 v = *(const float4*)(row + c4 * 4);
            // strict '>' in index order preserves first-occurrence argmax
            if (v.x > best) { best = v.x; bi = c4 * 4 + 0; }
            if (v.y > best) { best = v.y; bi = c4 * 4 + 1; }
            if (v.z > best) { best = v.z; bi = c4 * 4 + 2; }
            if (v.w > best) { best = v.w; bi = c4 * 4 + 3; }
        }
        atomicAdd(&lcounts[bi], 1);
    }
    __syncthreads();
    if (threadIdx.x < CCH) {
        int v = lcounts[threadIdx.x];
        if (v) atomicAdd(&counts[threadIdx.x], v);
    }
}

// ---- edge hinge: one wave handles 16 edges per group via WMMA Gram trick ----
__global__ __launch_bounds__(256) void edge_hinge_kernel(
    const float* __restrict__ out, const int* __restrict__ idx,
    long long E, float* __restrict__ partials) {
    const int lane   = threadIdx.x & 31;
    const int e_loc  = lane & 15;        // edge within group (M/N index)
    const int hi     = lane >> 4;        // 0: channels 0..31, 1: channels 32..63
    const int wave   = (int)((blockIdx.x * blockDim.x + threadIdx.x) >> 5);
    const int nwaves = (int)((gridDim.x * blockDim.x) >> 5);

    const long long ngroups = E >> 4;    // 16 edges per group
    float hinge = 0.0f;                  // identical across all 32 lanes
    float tail_hinge = 0.0f;             // per-lane scalar tail

    for (long long g = wave; g < ngroups; g += nwaves) {   // wave-uniform
        const long long e = (g << 4) + e_loc;
        const long long i = idx[e];
        const long long j = idx[E + e];
        const float* rowA = out + i * CCH + hi * 32;
        const float* rowB = out + j * CCH + hi * 32;

        v8f acc = {0.f, 0.f, 0.f, 0.f, 0.f, 0.f, 0.f, 0.f};
        #pragma unroll
        for (int it = 0; it < 8; ++it) {
            float4 a = *(const float4*)(rowA + it * 4);
            float4 b = *(const float4*)(rowB + it * 4);
            // A regs: .x -> K slot 0 (low lanes) / K slot 2 (high lanes)
            //         .y -> K slot 1 (low lanes) / K slot 3 (high lanes)
            v2f A0, A1;
            A0.x = a.x - b.x + EPSF;  A0.y = a.y - b.y + EPSF;
            A1.x = a.z - b.z + EPSF;  A1.y = a.w - b.w + EPSF;
            // D += A*A^T  (B layout == A layout for f32 16x16x4)
            acc = __builtin_amdgcn_wmma_f32_16x16x4_f32(
                false, A0, false, A0, (short)0, acc, false, false);
            acc = __builtin_amdgcn_wmma_f32_16x16x4_f32(
                false, A1, false, A1, (short)0, acc, false, false);
        }
        // Diagonal: edge m<8  -> acc[m]   at lane m
        //           edge m>=8 -> acc[m-8] at lane 24+(m-8)
        #pragma unroll
        for (int r = 0; r < 8; ++r) {
            float dlo = __shfl(acc[r], r, 32);
            float dhi = __shfl(acc[r], 24 + r, 32);
            hinge += fmaxf(MARGIN - sqrtf(dlo), 0.0f);
            hinge += fmaxf(MARGIN - sqrtf(dhi), 0.0f);
        }
    }

    // scalar tail for E % 16 != 0 (no WMMA; divergence is fine here)
    if (wave == 0) {
        long long e = (ngroups << 4) + lane;
        if (e < E) {
            const long long i = idx[e];
            const long long j = idx[E + e];
            const float* rowA = out + i * CCH;
            const float* rowB = out + j * CCH;
            float s = 0.0f;
            #pragma unroll
            for (int c = 0; c < CCH; ++c) {
                float d = rowA[c] - rowB[c] + EPSF;
                s = fmaf(d, d, s);
            }
            tail_hinge = fmaxf(MARGIN - sqrtf(s), 0.0f);
        }
    }

    // block reduction (fixed order -> deterministic partials)
    __shared__ float red[256];
    red[threadIdx.x] = ((lane == 0) ? hinge : 0.0f) + tail_hinge;
    __syncthreads();
    for (int s = 128; s > 0; s >>= 1) {
        if ((int)threadIdx.x < s) red[threadIdx.x] += red[threadIdx.x + s];
        __syncthreads();
    }
    if (threadIdx.x == 0) partials[blockIdx.x] = red[0];
}

// ---- final fixed-order reduction + log term ----
__global__ void finalize_kernel(const float* __restrict__ partials, int nparts,
                                const int* __restrict__ counts,
                                float* __restrict__ outv) {
    if (threadIdx.x == 0 && blockIdx.x == 0) {
        float h = 0.0f;
        for (int i = 0; i < nparts; ++i) h += partials[i];
        float lp = logf(0.1f);
        for (int c = 0; c < CCH; ++c) lp += logf((float)counts[c]);
        outv[0] = h - lp;
    }
}

extern "C" void kernel_launch(void* const* d_in, const int* in_sizes, int n_in,
                              void* d_out, int out_size, void* d_ws, size_t ws_size,
                              hipStream_t stream) {
    const float* output    = (const float*)d_in[0];  // [N, 64] f32
    const int*   edgeindex = (const int*)d_in[1];    // [2, E] i32
    const int       N = in_sizes[0] / CCH;
    const long long E = (long long)in_sizes[1] / 2;
    float* outp = (float*)d_out;

    int*   counts   = (int*)d_ws;                         // 64 ints
    float* partials = (float*)((char*)d_ws + 256);        // EDGE_BLOCKS floats

    init_counts_kernel<<<1, 64, 0, stream>>>(counts);
    argmax_count_kernel<<<(N + 255) / 256, 256, 0, stream>>>(output, N, counts);
    edge_hinge_kernel<<<EDGE_BLOCKS, 256, 0, stream>>>(output, edgeindex, E, partials);
    finalize_kernel<<<1, 32, 0, stream>>>(partials, EDGE_BLOCKS, counts, outp);
}